// EdgeDecoder_7765300871784
// MI455X (gfx1250) — compile-verified
//
#include <hip/hip_runtime.h>

typedef __attribute__((ext_vector_type(2))) float v2f;
typedef __attribute__((ext_vector_type(8))) float v8f;

#define H 128
#define TWO_H 256
#define EDGES_PER_BLOCK 128
#define W1_STRIDE 136   // 128 + 8 pad: consecutive k-rows start 8 banks apart
#define Z1_STRIDE 260   // 256 + 4 pad: edge rows start 4 banks apart (16 edges -> 16 distinct banks)

__global__ __launch_bounds__(256)
void edge_decoder_kernel(const float* __restrict__ z_author,
                         const float* __restrict__ z_hotel,
                         const int*   __restrict__ row,
                         const int*   __restrict__ col,
                         const float* __restrict__ W1,
                         const float* __restrict__ b1,
                         const float* __restrict__ W2,
                         const float* __restrict__ b2,
                         float* __restrict__ out_z3,
                         float* __restrict__ out_z1,
                         float* __restrict__ out_z2,
                         int E)
{
    __shared__ float lds_w1[TWO_H * W1_STRIDE];          // ~136 KB
    __shared__ float lds_z1[EDGES_PER_BLOCK * Z1_STRIDE];// ~130 KB
    __shared__ float lds_acc[EDGES_PER_BLOCK];

    const int tid  = threadIdx.x;
    const int lane = tid & 31;
    const int wid  = tid >> 5;                // 8 waves, one 16-col N-slice each
    const int e0   = blockIdx.x * EDGES_PER_BLOCK;
    const int nedges = min(EDGES_PER_BLOCK, E - e0);

    if (tid < EDGES_PER_BLOCK) lds_acc[tid] = b2[0];

    // ---- Stage W1 [256 x 128] into padded LDS (row-major k x n) ----
    for (int idx = tid; idx < TWO_H * (H / 4); idx += 256) {
        int k  = idx >> 5;          // 0..255
        int n4 = idx & 31;          // 0..31 float4s per row
        float4 v = ((const float4*)(W1 + (size_t)k * H))[n4];
        *(float4*)&lds_w1[k * W1_STRIDE + n4 * 4] = v;
    }

    // ---- Gather z1 = [z_author[row] | z_hotel[col]] into LDS + write z1 output ----
    for (int i = wid; i < nedges; i += 8) {
        int e  = e0 + i;
        int ra = row[e];
        int ca = col[e];
        float4 va = ((const float4*)(z_author + (size_t)ra * H))[lane];
        float4 vh = ((const float4*)(z_hotel  + (size_t)ca * H))[lane];
        *(float4*)&lds_z1[i * Z1_STRIDE +       lane * 4] = va;
        *(float4*)&lds_z1[i * Z1_STRIDE + H +   lane * 4] = vh;
        float4* po = (float4*)(out_z1 + (size_t)e * TWO_H);
        po[lane]      = va;
        po[lane + 32] = vh;
    }

    __syncthreads();

    // ---- Hoist this wave's B fragments (W1 columns n0..n0+15, all K) into registers ----
    // f32 WMMA B layout (4x16): VGPR0 = K rows {k, k+2} (lane halves), VGPR1 = {k+1, k+3}
    const int n0   = wid * 16;
    const int nn   = n0 + (lane & 15);
    const int koff = (lane >> 4) << 1;        // lanes 0-15 -> K+0/K+1 ; lanes 16-31 -> K+2/K+3
    v2f bfrag[64];
#pragma unroll
    for (int kk = 0; kk < 64; ++kk) {
        int k = kk * 4 + koff;
        v2f b;
        b.x = lds_w1[(k    ) * W1_STRIDE + nn];
        b.y = lds_w1[(k + 1) * W1_STRIDE + nn];
        bfrag[kk] = b;
    }

    const float w2n = W2[nn];
    const float b1n = b1[nn];
    const int nsub = nedges >> 4;             // E is a multiple of 16

    for (int s = 0; s < nsub; ++s) {
        // A layout (16x4 f32): lane m holds row m; VGPR0 = K {k,k+2}, VGPR1 = {k+1,k+3}
        const float* arow = &lds_z1[(s * 16 + (lane & 15)) * Z1_STRIDE + koff];
        v8f c = {};
#pragma unroll
        for (int kk = 0; kk < 64; ++kk) {
            v2f a = *(const v2f*)(arow + kk * 4);
            c = __builtin_amdgcn_wmma_f32_16x16x4_f32(false, a, false, bfrag[kk],
                                                      (short)0, c, false, false);
        }

        // C/D layout: VGPR r, lanes 0-15 -> M=r (N=lane), lanes 16-31 -> M=r+8 (N=lane-16)
        const int mbase = s * 16 + ((lane >> 4) << 3);
#pragma unroll
        for (int r = 0; r < 8; ++r) {
            float v = fmaxf(c[r] + b1n, 0.0f);
            int e = mbase + r;
            out_z2[(size_t)(e0 + e) * H + nn] = v;     // z2 output
            // z3 partial: reduce v * W2[n] over the 16 lanes of this M row
            float p = v * w2n;
            p += __shfl_xor(p, 1, 32);
            p += __shfl_xor(p, 2, 32);
            p += __shfl_xor(p, 4, 32);
            p += __shfl_xor(p, 8, 32);
            if ((lane & 15) == 0) atomicAdd(&lds_acc[e], p);   // ds_add_f32
        }
    }

    __syncthreads();
    if (tid < nedges) out_z3[e0 + tid] = lds_acc[tid];
}

extern "C" void kernel_launch(void* const* d_in, const int* in_sizes, int n_in,
                              void* d_out, int out_size, void* d_ws, size_t ws_size,
                              hipStream_t stream) {
    const float* z_author = (const float*)d_in[0];
    const float* z_hotel  = (const float*)d_in[1];
    const int*   row      = (const int*)d_in[2];
    const int*   col      = (const int*)d_in[3];
    const float* W1       = (const float*)d_in[4];
    const float* b1       = (const float*)d_in[5];
    const float* W2       = (const float*)d_in[6];
    const float* b2       = (const float*)d_in[7];
    const int E = in_sizes[2];

    float* out    = (float*)d_out;
    float* out_z3 = out;                                   // [E]
    float* out_z1 = out + (size_t)E;                       // [E, 256]
    float* out_z2 = out + (size_t)E + (size_t)E * TWO_H;   // [E, 128]

    int blocks = (E + EDGES_PER_BLOCK - 1) / EDGES_PER_BLOCK;
    edge_decoder_kernel<<<blocks, 256, 0, stream>>>(z_author, z_hotel, row, col,
                                                    W1, b1, W2, b2,
                                                    out_z3, out_z1, out_z2, E);
}